// MatchSegmentation_34067680591990
// MI455X (gfx1250) — compile-verified
//
#include <hip/hip_runtime.h>

typedef float v2f __attribute__((ext_vector_type(2)));
typedef float v8f __attribute__((ext_vector_type(8)));

#define KTOT 21
#define EPS  1e-6f
#define BIG  1.0e6f
#define WPB  8             // waves per block (wave32)
#define TPB  (WPB * 32)
#define PART 1088          // floats per wave partial: 4 tiles * 32 lanes * 8 + 64 (S1)

// ---------------- Phase 1: streaming WMMA GEMM partials -------------------
// C_tile[16x16] accumulates A(logdiff)[16 x n4] * B(gt)[n4 x 16] with
// V_WMMA_F32_16X16X4_F32. Two k-tiles (0-15, 16-20 pad) x two g-tiles.
// All loads are unconditional (clamped addresses); padding handled by
// branchless v_cndmask on the loaded values -> no exec-mask divergence.
__global__ __launch_bounds__(TPB)
void ce_phase1(const float* __restrict__ seg, const int* __restrict__ gt,
               float* __restrict__ ws, int N, int Gtot, int nwaves)
{
    const int lane = threadIdx.x & 31;
    const int w    = blockIdx.x * WPB + (threadIdx.x >> 5);
    const int lr   = lane & 15;
    const int half = lane >> 4;

    v8f c00 = {}; v8f c01 = {}; v8f c10 = {}; v8f c11 = {};
    float s1a = 0.f, s1b = 0.f;

    const int nstep = N >> 2;                        // 4 n-values per step
    const int chunk = (nstep + nwaves - 1) / nwaves; // contiguous chunk per wave
    const int sBeg  = w * chunk;
    const int sEnd  = (sBeg + chunk < nstep) ? (sBeg + chunk) : nstep;

    const bool kv1 = (16 + lr) < KTOT;               // second k-tile row valid
    const bool gv1 = (16 + lr) < Gtot;               // second g-tile col valid
    const int  kc1 = kv1 ? (16 + lr) : (KTOT - 1);   // clamped safe seg column
    const int  gc1 = gv1 ? (16 + lr) : (Gtot - 1);   // clamped safe gt row

    // Base pointers for this wave's contiguous chunk; bumped by fixed strides.
    const float* segp = seg + (size_t)((size_t)sBeg * 4 + half * 2) * KTOT;
    const int*   gtp0 = gt + (size_t)lr  * N + (size_t)sBeg * 4 + half * 2;
    const int*   gtp1 = gt + (size_t)gc1 * N + (size_t)sBeg * 4 + half * 2;

    #pragma unroll 4
    for (int s = sBeg; s < sEnd; ++s) {
        // ---- A: seg values for k = lr (tile0) and clamped 16+lr (tile1)
        float x0 = segp[lr];
        float x1 = segp[KTOT + lr];
        float y0 = segp[kc1];
        float y1 = segp[KTOT + kc1];
        // ---- B: two contiguous gt values per row -> single b64 load
        int2 gb0 = *(const int2*)gtp0;
        int2 gb1 = *(const int2*)gtp1;

        float lx0p = __logf(x0 + EPS), lx0m = __logf(1.f - x0 + EPS);
        float lx1p = __logf(x1 + EPS), lx1m = __logf(1.f - x1 + EPS);
        float ly0p = __logf(y0 + EPS), ly0m = __logf(1.f - y0 + EPS);
        float ly1p = __logf(y1 + EPS), ly1m = __logf(1.f - y1 + EPS);

        v2f a0, a1;
        a0.x = lx0p - lx0m;                          // VGPR0 = n   (K=0 / K=2)
        a0.y = lx1p - lx1m;                          // VGPR1 = n+1 (K=1 / K=3)
        a1.x = kv1 ? (ly0p - ly0m) : 0.f;            // branchless pad
        a1.y = kv1 ? (ly1p - ly1m) : 0.f;

        s1a += lx0m + lx1m;                          // S1 bias, k = lr
        s1b += kv1 ? (ly0m + ly1m) : 0.f;            // S1 bias, k = 16+lr

        v2f b0, b1;
        b0.x = (float)gb0.x;
        b0.y = (float)gb0.y;
        b1.x = gv1 ? (float)gb1.x : 0.f;             // branchless pad
        b1.y = gv1 ? (float)gb1.y : 0.f;

        // D = A*B + C  (f32 WMMA: exact f32 accumulation, matches jnp)
        c00 = __builtin_amdgcn_wmma_f32_16x16x4_f32(false, a0, false, b0, (short)0, c00, false, false);
        c01 = __builtin_amdgcn_wmma_f32_16x16x4_f32(false, a0, false, b1, (short)0, c01, false, false);
        c10 = __builtin_amdgcn_wmma_f32_16x16x4_f32(false, a1, false, b0, (short)0, c10, false, false);
        c11 = __builtin_amdgcn_wmma_f32_16x16x4_f32(false, a1, false, b1, (short)0, c11, false, false);

        segp += 4 * KTOT;
        gtp0 += 4;
        gtp1 += 4;
    }

    // Raw lane-major partial: ws[w][tile(4)][lane(32)][r(8)] + S1[2][32].
    // Every wave writes (zero-init accumulators) -> ws fully defined each call.
    float* p  = ws + (size_t)w * PART;
    v8f*   pv = (v8f*)p;
    pv[0 * 32 + lane] = c00;   // (k-tile0, g-tile0)
    pv[1 * 32 + lane] = c01;   // (k-tile0, g-tile1)
    pv[2 * 32 + lane] = c10;   // (k-tile1, g-tile0)
    pv[3 * 32 + lane] = c11;   // (k-tile1, g-tile1)
    p[1024 + lane]      = s1a;
    p[1024 + 32 + lane] = s1b;
}

// ---------------- Phase 2: fixed-order reduction + matching epilogue ------
__global__ __launch_bounds__(256)
void ce_phase2(const float* __restrict__ ws, const int* __restrict__ gpn,
               float* __restrict__ out, int N, int Gtot, int nwaves)
{
    __shared__ float C[1024];
    __shared__ float S[64];
    const int t = threadIdx.x;
    for (int idx = t; idx < PART; idx += 256) {
        float acc = 0.f;
        for (int w = 0; w < nwaves; ++w)           // fixed order -> deterministic
            acc += ws[(size_t)w * PART + idx];
        if (idx < 1024) C[idx] = acc; else S[idx - 1024] = acc;
    }
    __syncthreads();

    if (t == 0) {
        int G = *gpn;
        if (G < 1) G = 1;
        if (G > Gtot) G = Gtot;
        const float invN = 1.0f / (float)N;

        int   matching[KTOT];
        float ceval[KTOT];
        for (int k = 0; k < KTOT; ++k) {
            const int kt = k >> 4, m = k & 15;
            const int h = m >> 3, r = m & 7;
            const float s1k = (kt == 0) ? (S[k] + S[k + 16])
                                        : (S[32 + (k - 16)] + S[32 + (k - 16) + 16]);
            int bm = -1; float bv = 0.f;
            for (int g = 0; g < G; ++g) {          // first-min argmin (jnp semantics)
                const int gtile = g >> 4, col = g & 15;
                const int lanei = h * 16 + col;
                const float dot = C[(kt * 2 + gtile) * 256 + lanei * 8 + r];
                const float ce  = -(dot + s1k) * invN;
                if (bm < 0 || ce < bv) { bv = ce; bm = g; }
            }
            matching[k] = bm; ceval[k] = bv;
        }

        int maxidx = 0;
        for (int k = 0; k < KTOT; ++k) maxidx = (matching[k] > maxidx) ? matching[k] : maxidx;
        maxidx += 1;

        int bestk[32];
        for (int g = 0; g < G; ++g) {
            float bv = __builtin_inff(); int bk = 0;
            for (int k = 0; k < KTOT; ++k) {       // first-min; all-BIG row -> 0
                const float v = (matching[k] == g) ? ceval[k] : BIG;
                if (v < bv) { bv = v; bk = k; }
            }
            bestk[g] = bk;
        }
        for (int k = 0; k < KTOT; ++k) {
            const int m = matching[k];
            out[k] = (float)((bestk[m] == k) ? m : maxidx);
        }
    }
}

extern "C" void kernel_launch(void* const* d_in, const int* in_sizes, int n_in,
                              void* d_out, int out_size, void* d_ws, size_t ws_size,
                              hipStream_t stream)
{
    const float* seg = (const float*)d_in[0];
    const int*   gt  = (const int*)d_in[1];
    const int*   gpn = (const int*)d_in[2];

    const int N = in_sizes[0] / KTOT;      // 230400
    int Gtot    = in_sizes[1] / N;         // 21
    if (Gtot > 32) Gtot = 32;

    // Clamp wave count by scratch capacity (deterministic: depends only on ws_size).
    const int maxWaves = (int)(ws_size / ((size_t)PART * sizeof(float)));
    int blocks = 120;
    if (blocks * WPB > maxWaves) blocks = maxWaves / WPB;
    if (blocks < 1) blocks = 1;
    const int nwaves = blocks * WPB;

    float* ws = (float*)d_ws;
    ce_phase1<<<blocks, TPB, 0, stream>>>(seg, gt, ws, N, Gtot, nwaves);
    ce_phase2<<<1, 256, 0, stream>>>(ws, gpn, (float*)d_out, N, Gtot, nwaves);
}